// AnnealingTopKSoftMax_56392920597027
// MI455X (gfx1250) — compile-verified
//
#include <hip/hip_runtime.h>
#include <stdint.h>

// ---------------------------------------------------------------------------
// Top-K(8) + softmax over rows of a [131072, 512] f32 matrix.
// One wave32 per row; 8 waves (256 threads) per block.
// CDNA5 path: async global->LDS staging (ASYNCcnt) + NT b128 stores.
// ---------------------------------------------------------------------------

#define DDIM 512
#define KSEL 8
#define WPB  8             // waves per block
#define TPB  (WPB * 32)

#if defined(__has_builtin)
#  if __has_builtin(__builtin_amdgcn_global_load_async_to_lds_b128) && \
      __has_builtin(__builtin_amdgcn_s_wait_asynccnt)
#    define USE_ASYNC_LDS 1
#  endif
#endif
#ifndef USE_ASYNC_LDS
#  define USE_ASYNC_LDS 0
#endif

// native clang vector types (HIP float4 is a class -> rejected by NT builtins)
typedef float v4f __attribute__((ext_vector_type(4)));
typedef int   v4i __attribute__((ext_vector_type(4)));
typedef __attribute__((address_space(1))) v4i* as1_v4i;
typedef __attribute__((address_space(3))) v4i* as3_v4i;

// descending compare-exchange: a <- max, b <- min
__device__ __forceinline__ void ce_desc(float& a, float& b) {
    float hi = fmaxf(a, b);
    float lo = fminf(a, b);
    a = hi; b = lo;
}

// Batcher odd-even mergesort, 8 elements, descending. 19 comparators.
__device__ __forceinline__ void sort8_desc(float t[8]) {
    ce_desc(t[0], t[1]); ce_desc(t[2], t[3]); ce_desc(t[4], t[5]); ce_desc(t[6], t[7]);
    ce_desc(t[0], t[2]); ce_desc(t[1], t[3]); ce_desc(t[4], t[6]); ce_desc(t[5], t[7]);
    ce_desc(t[1], t[2]); ce_desc(t[5], t[6]);
    ce_desc(t[0], t[4]); ce_desc(t[1], t[5]); ce_desc(t[2], t[6]); ce_desc(t[3], t[7]);
    ce_desc(t[2], t[4]); ce_desc(t[3], t[5]);
    ce_desc(t[1], t[2]); ce_desc(t[3], t[4]); ce_desc(t[5], t[6]);
}

// Bitonic cleaner for 8 (input bitonic, output sorted descending). 12 comparators.
__device__ __forceinline__ void bitonic_merge8_desc(float t[8]) {
    ce_desc(t[0], t[4]); ce_desc(t[1], t[5]); ce_desc(t[2], t[6]); ce_desc(t[3], t[7]);
    ce_desc(t[0], t[2]); ce_desc(t[1], t[3]); ce_desc(t[4], t[6]); ce_desc(t[5], t[7]);
    ce_desc(t[0], t[1]); ce_desc(t[2], t[3]); ce_desc(t[4], t[5]); ce_desc(t[6], t[7]);
}

__global__ __launch_bounds__(TPB)
void AnnealingTopKSoftMax_56392920597027_kernel(const float* __restrict__ in,
                                                float* __restrict__ out,
                                                int rows) {
    __shared__ __align__(16) float lds[WPB][DDIM];

    const int lane = threadIdx.x & 31;
    const int wv   = threadIdx.x >> 5;
    const int row  = blockIdx.x * WPB + wv;
    if (row >= rows) return;

    const float* g = in + (size_t)row * DDIM;

    // ---- load 16 elements per lane: slot s = c*4+j  -> global col c*128 + lane*4 + j
    float f[16];
#if USE_ASYNC_LDS
    float* l = &lds[wv][0];
    #pragma unroll
    for (int c = 0; c < 4; ++c) {
        __builtin_amdgcn_global_load_async_to_lds_b128(
            (as1_v4i)(uintptr_t)(g + c * 128 + lane * 4),
            (as3_v4i)(uintptr_t)(l + c * 128 + lane * 4),
            0, 0);
    }
    __builtin_amdgcn_s_wait_asynccnt(0);
    asm volatile("" ::: "memory");   // compiler barrier: LDS now holds the row
    #pragma unroll
    for (int c = 0; c < 4; ++c) {
        v4f v = *(const v4f*)(l + c * 128 + lane * 4);
        f[c * 4 + 0] = v.x; f[c * 4 + 1] = v.y;
        f[c * 4 + 2] = v.z; f[c * 4 + 3] = v.w;
    }
#else
    #pragma unroll
    for (int c = 0; c < 4; ++c) {
        v4f v = __builtin_nontemporal_load((const v4f*)(g + c * 128 + lane * 4));
        f[c * 4 + 0] = v.x; f[c * 4 + 1] = v.y;
        f[c * 4 + 2] = v.z; f[c * 4 + 3] = v.w;
    }
#endif

    // ---- lane-local sorted top-8 of 16 (value multiset; no divergence)
    float a[8], b[8];
    #pragma unroll
    for (int i = 0; i < 8; ++i) { a[i] = f[i]; b[i] = f[8 + i]; }
    sort8_desc(a);
    sort8_desc(b);
    float t[8];
    #pragma unroll
    for (int i = 0; i < 8; ++i) t[i] = fmaxf(a[i], b[7 - i]);
    bitonic_merge8_desc(t);

    // ---- butterfly merge across the wave: every lane ends with global top-8 sorted
    #pragma unroll
    for (int d = 1; d < 32; d <<= 1) {
        float o[8];
        #pragma unroll
        for (int i = 0; i < 8; ++i) o[i] = __shfl_xor(t[i], d, 32);
        #pragma unroll
        for (int i = 0; i < 8; ++i) t[i] = fmaxf(t[i], o[7 - i]);
        bitonic_merge8_desc(t);
    }

    const float rowmax = t[0];
    const float thr    = t[7];          // 8th-largest value (multiset semantics)

    // Exact denominator from the top-8 value multiset (tie-break independent).
    float denom = 0.0f;
    #pragma unroll
    for (int i = 0; i < 8; ++i) denom += __expf(t[i] - rowmax);
    const float inv = 1.0f / denom;

    // ---- which indices are selected (match jax.lax.top_k: ties -> lowest index)
    int packed = 0;                      // (count_gt << 16) | count_eq, totals <= 512
    #pragma unroll
    for (int s = 0; s < 16; ++s) {
        packed += (f[s] > thr) ? (1 << 16) : 0;
        packed += (f[s] == thr) ? 1 : 0;
    }
    #pragma unroll
    for (int off = 16; off; off >>= 1) packed += __shfl_xor(packed, off, 32);
    const int cnt_gt = packed >> 16;
    const int cnt_eq = packed & 0xffff;
    const int need   = KSEL - cnt_gt;    // >= 1 by construction of thr

    unsigned selMask = 0;
    #pragma unroll
    for (int s = 0; s < 16; ++s)
        if (f[s] > thr) selMask |= (1u << s);

    if (cnt_eq == need) {
        // common case: boundary value unique (or all ties fit) -> take all equals
        #pragma unroll
        for (int s = 0; s < 16; ++s)
            if (f[s] == thr) selMask |= (1u << s);
    } else {
        // astronomically rare: multiple equals straddle the boundary;
        // pick 'need' of them with the smallest global indices (wave-uniform loop).
        int remaining = need;
        while (remaining > 0) {
            int best = 0x7fffffff;
            #pragma unroll
            for (int s = 0; s < 16; ++s) {
                if ((f[s] == thr) && !((selMask >> s) & 1u)) {
                    int gi = ((s >> 2) << 7) + (lane << 2) + (s & 3);
                    best = min(best, gi);
                }
            }
            #pragma unroll
            for (int off = 16; off; off >>= 1) best = min(best, __shfl_xor(best, off, 32));
            #pragma unroll
            for (int s = 0; s < 16; ++s) {
                int gi = ((s >> 2) << 7) + (lane << 2) + (s & 3);
                if (gi == best && f[s] == thr) selMask |= (1u << s);
            }
            --remaining;
        }
    }

    // ---- write softmax: selected -> exp(v-max)/denom, else exact 0 (NT b128 stores)
    float* orow = out + (size_t)row * DDIM;
    #pragma unroll
    for (int c = 0; c < 4; ++c) {
        v4f o;
        o.x = ((selMask >> (c * 4 + 0)) & 1u) ? __expf(f[c * 4 + 0] - rowmax) * inv : 0.0f;
        o.y = ((selMask >> (c * 4 + 1)) & 1u) ? __expf(f[c * 4 + 1] - rowmax) * inv : 0.0f;
        o.z = ((selMask >> (c * 4 + 2)) & 1u) ? __expf(f[c * 4 + 2] - rowmax) * inv : 0.0f;
        o.w = ((selMask >> (c * 4 + 3)) & 1u) ? __expf(f[c * 4 + 3] - rowmax) * inv : 0.0f;
        __builtin_nontemporal_store(o, (v4f*)(orow + c * 128 + lane * 4));
    }
}

extern "C" void kernel_launch(void* const* d_in, const int* in_sizes, int n_in,
                              void* d_out, int out_size, void* d_ws, size_t ws_size,
                              hipStream_t stream) {
    const float* in  = (const float*)d_in[0];
    float*       out = (float*)d_out;
    const int rows   = in_sizes[0] / DDIM;          // 131072
    const int blocks = (rows + WPB - 1) / WPB;      // 16384
    hipLaunchKernelGGL(AnnealingTopKSoftMax_56392920597027_kernel,
                       dim3(blocks), dim3(TPB), 0, stream, in, out, rows);
}